// OrthoLoss_6055903887416
// MI455X (gfx1250) — compile-verified
//
#include <hip/hip_runtime.h>
#include <hip/hip_bf16.h>
#include <math.h>

typedef __attribute__((ext_vector_type(2))) float v2f;
typedef __attribute__((ext_vector_type(8))) float v8f;

#define NB   32      // batches
#define NS   2048    // rows per batch
#define ND   128     // feature dim
#define NSEG 8       // row segments per (batch, col-tile)
#define ROWS_PER_SEG (NS / NSEG)   // 256
#define EPSN 1e-8f

// ---------------------------------------------------------------------------
// Kernel 1: per-row inverse clamped norms. One wave (32 lanes) per row,
// each lane loads a float4 (16B) -> 512B coalesced per row.
// ---------------------------------------------------------------------------
__global__ void ortho_norms_kernel(const float* __restrict__ x,
                                   float* __restrict__ w) {
    const int row  = blockIdx.x * 8 + (threadIdx.x >> 5);  // 8 waves / block
    const int lane = threadIdx.x & 31;

    const float4* xr = (const float4*)(x + (size_t)row * ND);
    float4 p = xr[lane];
    float ss = p.x * p.x + p.y * p.y + p.z * p.z + p.w * p.w;

    // wave32 butterfly reduction
    #pragma unroll
    for (int m = 16; m >= 1; m >>= 1)
        ss += __shfl_xor(ss, m, 32);

    if (lane == 0)
        w[row] = 1.0f / fmaxf(sqrtf(ss), EPSN);
}

// ---------------------------------------------------------------------------
// Kernel 2: v_b = X^T w via V_WMMA_F32_16X16X4_F32 (fp32 WMMA, no precision
// loss). One wave per (batch, 16-col tile, row segment).
//   A(16x4): A[m,k] = w[s0+k]          (broadcast over M; lane-half = K pair)
//   B(4x16): B[k,n] = x[s0+k, d0+n]
//   D rows all equal the accumulated column sums v[d0..d0+15].
// blockDim = 32, no divergence before WMMA -> EXEC all ones as required.
// ---------------------------------------------------------------------------
__global__ void ortho_wmma_accum_kernel(const float* __restrict__ x,
                                        const float* __restrict__ w,
                                        float* __restrict__ vpart) {
    const int wid  = blockIdx.x;               // 0 .. NB*8*NSEG-1
    const int b    = wid / (8 * NSEG);
    const int rem  = wid % (8 * NSEG);
    const int tile = rem / NSEG;               // 0..7
    const int seg  = rem % NSEG;               // 0..NSEG-1
    const int d0   = tile * 16;

    const int lane = threadIdx.x;              // 0..31
    const int n    = lane & 15;                // column within tile
    const int h    = lane >> 4;                // lane half -> K pair {2h, 2h+1}

    const float* xb = x + (size_t)b * NS * ND;
    const float* wb = w + (size_t)b * NS;

    v8f c = {};
    const int sbase = seg * ROWS_PER_SEG;
    #pragma unroll 4
    for (int s0 = sbase; s0 < sbase + ROWS_PER_SEG; s0 += 4) {
        const int r = s0 + 2 * h;
        // A: lanes 0-15 -> {w[s0], w[s0+1]}, lanes 16-31 -> {w[s0+2], w[s0+3]}
        v2f a = *(const v2f*)(wb + r);         // r even -> 8B aligned
        // B: same K indexing convention, column d0+n
        v2f bb;
        bb.x = xb[(size_t)r * ND + d0 + n];
        bb.y = xb[(size_t)(r + 1) * ND + d0 + n];
        // 8 args: (neg_a, A, neg_b, B, c_mod, C, reuse_a, reuse_b)
        c = __builtin_amdgcn_wmma_f32_16x16x4_f32(
                false, a, false, bb, (short)0, c, false, false);
    }

    // D VGPR0, lanes 0-15 = row M=0, N=lane -> the partial column sums.
    if (lane < 16)
        vpart[((size_t)seg * NB + b) * ND + d0 + n] = c[0];
}

// ---------------------------------------------------------------------------
// Kernel 3: v[i] = sum_seg vpart[seg][i] (fixed order -> deterministic),
// result = sum_i v[i]^2 / (NB*NS*NS).
// ---------------------------------------------------------------------------
__global__ void ortho_finalize_kernel(const float* __restrict__ vpart,
                                      float* __restrict__ out) {
    __shared__ float red[256];
    float acc = 0.0f;
    for (int i = threadIdx.x; i < NB * ND; i += 256) {
        float vi = 0.0f;
        #pragma unroll
        for (int seg = 0; seg < NSEG; ++seg)
            vi += vpart[(size_t)seg * NB * ND + i];
        acc += vi * vi;
    }
    red[threadIdx.x] = acc;
    __syncthreads();
    for (int k = 128; k >= 1; k >>= 1) {
        if (threadIdx.x < k) red[threadIdx.x] += red[threadIdx.x + k];
        __syncthreads();
    }
    if (threadIdx.x == 0)
        out[0] = red[0] * (1.0f / 134217728.0f);   // 1 / (32*2048*2048) = 2^-27
}

// ---------------------------------------------------------------------------
extern "C" void kernel_launch(void* const* d_in, const int* in_sizes, int n_in,
                              void* d_out, int out_size, void* d_ws, size_t ws_size,
                              hipStream_t stream) {
    const float* x = (const float*)d_in[0];
    float* out = (float*)d_out;

    float* w     = (float*)d_ws;          // NB*NS floats (256 KB)
    float* vpart = w + NB * NS;           // NSEG*NB*ND floats (128 KB)

    // 1) inverse clamped row norms: 65536 rows, 8 rows per 256-thread block
    ortho_norms_kernel<<<(NB * NS) / 8, 256, 0, stream>>>(x, w);

    // 2) WMMA column-sum accumulation: one wave per (b, tile, seg)
    ortho_wmma_accum_kernel<<<NB * 8 * NSEG, 32, 0, stream>>>(x, w, vpart);

    // 3) deterministic segment reduction + squared-norm + scale
    ortho_finalize_kernel<<<1, 256, 0, stream>>>(vpart, out);
}